// ECELoss_25323127177868
// MI455X (gfx1250) — compile-verified
//
#include <hip/hip_runtime.h>
#include <cstdint>

#define TPB   256
#define NCLS  10
#define NBINS 10
#define SPT   512                               // samples per tile
#define TILE_BYTES (SPT * NCLS * 4)             // 20480 B
#define CHUNKS ((TILE_BYTES / 16) / TPB)        // 5 x b128 per thread
static_assert(CHUNKS * TPB * 16 == TILE_BYTES, "tile must tile evenly");

// gfx1250 async copy engine, 64-bit vaddr form (clamped tail path):
//   copies 16B per lane from global vaddr to LDS address vdst
__device__ __forceinline__ void async_b128_gv(unsigned lds_addr, uint64_t gaddr) {
  asm volatile("global_load_async_to_lds_b128 %0, %1, off"
               :: "v"(lds_addr), "v"(gaddr) : "memory");
}

// GVS form (steady-state fast path): base in SGPR pair, 32-bit per-lane voff,
// compile-time chunk stride in the offset: immediate. Per ISA, INST_OFFSET is
// added to BOTH the LDS address (vdst + off) and global (saddr + voff + off),
// which matches our identical LDS/global tile layout.
template <int IMM>
__device__ __forceinline__ void async_b128_gvs(unsigned lds_addr, unsigned voff,
                                               uint64_t sbase) {
  asm volatile("global_load_async_to_lds_b128 %0, %1, %2 offset:%c3"
               :: "v"(lds_addr), "v"(voff), "s"(sbase), "n"(IMM)
               : "memory");
}

__global__ __launch_bounds__(TPB) void ece_hist(
    const float* __restrict__ probs,
    const int*   __restrict__ labels,
    int n, int numTiles,
    float*    __restrict__ g_conf,
    unsigned* __restrict__ g_cnt,
    unsigned* __restrict__ g_cor)
{
  __shared__ float    s_buf[2][SPT * NCLS];     // 2 x 20 KB staging buffers
  __shared__ float    s_conf[NBINS][32];        // per-(bin,lane): conflict-free
  __shared__ unsigned s_cnt[NBINS][32];
  __shared__ unsigned s_cor[NBINS][32];

  const int tid  = threadIdx.x;
  const int lane = tid & 31;
  const size_t totalBytes = (size_t)n * (NCLS * 4);
  const char* gbase = (const char*)probs;

  // Exactly CHUNKS async issues per thread on BOTH paths (uniform per wave) so
  // s_wait_asynccnt immediates stay correct. Only the last tile needs clamping.
  auto issue = [&](int tile, int bufIdx) {
    const size_t   tileBase = (size_t)tile * TILE_BYTES;
    const unsigned ldsBase  = (unsigned)(uintptr_t)(&s_buf[bufIdx][0]);
    if (tile != numTiles - 1) {                 // uniform branch: fast path
      const uint64_t sbase = (uint64_t)(uintptr_t)gbase + tileBase;
      const unsigned voff  = (unsigned)(tid * 16);
      const unsigned ldsA  = ldsBase + voff;
      async_b128_gvs<0 * TPB * 16>(ldsA, voff, sbase);
      async_b128_gvs<1 * TPB * 16>(ldsA, voff, sbase);
      async_b128_gvs<2 * TPB * 16>(ldsA, voff, sbase);
      async_b128_gvs<3 * TPB * 16>(ldsA, voff, sbase);
      async_b128_gvs<4 * TPB * 16>(ldsA, voff, sbase);
    } else {                                    // last tile: clamp OOB chunks
#pragma unroll
      for (int k = 0; k < CHUNKS; ++k) {
        const unsigned off = (unsigned)(tid * 16 + k * (TPB * 16));
        size_t g = tileBase + off;
        if (g + 16 > totalBytes) g = totalBytes - 16;   // dup of last 16B; never read
        async_b128_gv(ldsBase + off, (uint64_t)(uintptr_t)(gbase + g));
      }
    }
  };

  issue(blockIdx.x, 0);                         // prologue: first tile -> buf 0

  for (int i = tid; i < NBINS * 32; i += TPB) { // zero hist while copy flies
    (&s_conf[0][0])[i] = 0.0f;
    (&s_cnt[0][0])[i]  = 0u;
    (&s_cor[0][0])[i]  = 0u;
  }

  int parity = 0;
  for (int t = blockIdx.x; t < numTiles; t += (int)gridDim.x) {
    const int nxt = t + (int)gridDim.x;
    if (nxt < numTiles) {
      issue(nxt, parity ^ 1);                   // prefetch next tile
      asm volatile("s_wait_asynccnt 0x5" ::: "memory"); // drain current tile only
    } else {
      asm volatile("s_wait_asynccnt 0x0" ::: "memory");
    }
    __syncthreads();                            // all waves' copies landed

    const float* buf = s_buf[parity];
#pragma unroll
    for (int ss = 0; ss < SPT / TPB; ++ss) {
      const int  sl     = tid + ss * TPB;
      const long sample = (long)t * SPT + sl;
      if (sample < n) {
        float r[NCLS];
#pragma unroll
        for (int j = 0; j < NCLS; ++j) r[j] = buf[sl * NCLS + j];

        float best = r[0]; int arg = 0;         // first-max argmax (jnp semantics)
#pragma unroll
        for (int j = 1; j < NCLS; ++j)
          if (r[j] > best) { best = r[j]; arg = j; }

        const int  lab     = labels[sample];
        const bool correct = (arg == lab);

#pragma unroll
        for (int j = 0; j < NCLS; ++j) {
          const float x = r[j];
          // digitize(x, edges, right=True)-1 == #{e : x > e} - 1 ; x==0 -> -1 (drop)
          const float EDGES[11] = {0.0f,0.1f,0.2f,0.3f,0.4f,0.5f,
                                   0.6f,0.7f,0.8f,0.9f,1.0f};
          int idx = -1;
#pragma unroll
          for (int e = 0; e < 11; ++e) idx += (x > EDGES[e]) ? 1 : 0;
          if ((unsigned)idx < (unsigned)NBINS) {
            atomicAdd(&s_cnt[idx][lane], 1u);   // ds_add_u32
            atomicAdd(&s_conf[idx][lane], x);   // ds_add_f32
            if (correct && j == lab)
              atomicAdd(&s_cor[idx][lane], 1u);
          }
        }
      }
    }
    __syncthreads();                            // done reading buf before reissue
    parity ^= 1;
  }

  // Block reduction: 30 accumulators, 32 lanes each, then one global atomic.
  if (tid < 3 * NBINS) {
    const int a = tid / NBINS, b = tid % NBINS;
    if (a == 0) {
      float s = 0.0f;
      for (int l = 0; l < 32; ++l) s += s_conf[b][l];
      __hip_atomic_fetch_add(&g_conf[b], s, __ATOMIC_RELAXED,
                             __HIP_MEMORY_SCOPE_AGENT);
    } else if (a == 1) {
      unsigned s = 0u;
      for (int l = 0; l < 32; ++l) s += s_cnt[b][l];
      atomicAdd(&g_cnt[b], s);
    } else {
      unsigned s = 0u;
      for (int l = 0; l < 32; ++l) s += s_cor[b][l];
      atomicAdd(&g_cor[b], s);
    }
  }
}

__global__ void ece_init(float* g_conf, unsigned* g_cnt, unsigned* g_cor) {
  const int i = threadIdx.x;
  if (i < NBINS) { g_conf[i] = 0.0f; g_cnt[i] = 0u; g_cor[i] = 0u; }
}

__global__ void ece_final(const float* __restrict__ g_conf,
                          const unsigned* __restrict__ g_cnt,
                          const unsigned* __restrict__ g_cor,
                          float* __restrict__ out) {
  if (threadIdx.x == 0 && blockIdx.x == 0) {
    const float EDGES[11] = {0.0f,0.1f,0.2f,0.3f,0.4f,0.5f,
                             0.6f,0.7f,0.8f,0.9f,1.0f};
    float total = 0.0f, num = 0.0f;
    for (int b = 0; b < NBINS; ++b) {
      const float c    = (float)g_cnt[b];
      const float conf = g_conf[b] / c;
      const float acc  = (float)g_cor[b] / c;
      num   += fabsf(conf - acc) * c;
      total += c;
      out[1 + b]  = EDGES[b + 1] - 0.05f;       // bin centers
      out[11 + b] = acc;                        // bin accuracy
    }
    out[0] = num / total;                       // ece
  }
}

extern "C" void kernel_launch(void* const* d_in, const int* in_sizes, int n_in,
                              void* d_out, int out_size, void* d_ws, size_t ws_size,
                              hipStream_t stream) {
  const float* probs  = (const float*)d_in[0];   // [N, 10] f32
  const int*   labels = (const int*)d_in[1];     // [N] i32
  const int n = in_sizes[1];

  float*    g_conf = (float*)d_ws;
  unsigned* g_cnt  = (unsigned*)((char*)d_ws + 64);
  unsigned* g_cor  = (unsigned*)((char*)d_ws + 128);

  const int numTiles = (n + SPT - 1) / SPT;
  const int blocks   = numTiles < 1024 ? numTiles : 1024;

  hipLaunchKernelGGL(ece_init,  dim3(1), dim3(32), 0, stream, g_conf, g_cnt, g_cor);
  hipLaunchKernelGGL(ece_hist,  dim3(blocks), dim3(TPB), 0, stream,
                     probs, labels, n, numTiles, g_conf, g_cnt, g_cor);
  hipLaunchKernelGGL(ece_final, dim3(1), dim3(32), 0, stream,
                     g_conf, g_cnt, g_cor, (float*)d_out);
}